// GCN_REDDIT_65781719105727
// MI455X (gfx1250) — compile-verified
//
#include <hip/hip_runtime.h>
#include <math.h>

typedef __attribute__((ext_vector_type(2))) float v2f;
typedef __attribute__((ext_vector_type(8))) float v8f;

#define GCN_N 100000
#define GCN_FIN 602
#define GCN_H 256
#define GCN_C 42

// ---------------- degree kernels ----------------
__global__ void k_deg_init(float* deg, int n) {
    int i = blockIdx.x * blockDim.x + threadIdx.x;
    if (i < n) deg[i] = 1.0f;   // self loop
}

__global__ void k_deg_accum(const int* __restrict__ dst, float* deg, int e) {
    int i = blockIdx.x * blockDim.x + threadIdx.x;
    if (i < e) atomicAdd(&deg[dst[i]], 1.0f);
}

__global__ void k_deg_rsqrt(float* deg, int n) {
    int i = blockIdx.x * blockDim.x + threadIdx.x;
    if (i < n) deg[i] = rsqrtf(deg[i]);
}

// ---------------- GEMM1: [N,602] x [602,256] -> [N,256] via V_WMMA_F32_16X16X4_F32
__global__ void __launch_bounds__(256) k_gemm1_wmma(const float* __restrict__ A,
                                                    const float* __restrict__ B,
                                                    float* __restrict__ Out,
                                                    int n) {
    const int K = GCN_FIN;
    int wave = (blockIdx.x * blockDim.x + threadIdx.x) >> 5;
    int lane = threadIdx.x & 31;
    int row0 = wave * 16;
    if (row0 >= n) return;
    int kgrp = lane >> 4;      // 0: K pair {k,k+1}, 1: {k+2,k+3}
    int l16  = lane & 15;
    int m = row0 + l16;
    if (m >= n) m = n - 1;     // clamp (n % 16 == 0 in practice)
    const float* arow = A + (size_t)m * K;

    v8f z = {0.f,0.f,0.f,0.f,0.f,0.f,0.f,0.f};
    v8f c[16];
    #pragma unroll
    for (int t = 0; t < 16; ++t) c[t] = z;

    int k = 0;
    for (; k <= K - 4; k += 4) {            // 150 unguarded steps, K=0..599
        int kb = k + kgrp * 2;
        v2f a;
        a.x = arow[kb];
        a.y = arow[kb + 1];
        const float* brow = B + (size_t)kb * GCN_H;
        #pragma unroll
        for (int t = 0; t < 16; ++t) {
            int col = t * 16 + l16;
            v2f b;
            b.x = brow[col];
            b.y = brow[GCN_H + col];
            c[t] = __builtin_amdgcn_wmma_f32_16x16x4_f32(
                false, a, false, b, (short)0, c[t], false, false);
        }
    }
    {   // tail: k = 600, elements 600..601 valid, 602..603 zero-padded
        int kb = k + kgrp * 2;
        bool ok0 = (kb     < K);
        bool ok1 = (kb + 1 < K);
        v2f a;
        a.x = ok0 ? arow[kb]     : 0.f;
        a.y = ok1 ? arow[kb + 1] : 0.f;
        #pragma unroll
        for (int t = 0; t < 16; ++t) {
            int col = t * 16 + l16;
            v2f b;
            b.x = ok0 ? B[(size_t)kb * GCN_H + col]       : 0.f;
            b.y = ok1 ? B[(size_t)(kb + 1) * GCN_H + col] : 0.f;
            c[t] = __builtin_amdgcn_wmma_f32_16x16x4_f32(
                false, a, false, b, (short)0, c[t], false, false);
        }
    }
    int rbase = row0 + kgrp * 8;
    #pragma unroll
    for (int t = 0; t < 16; ++t) {
        int col = t * 16 + l16;
        #pragma unroll
        for (int r = 0; r < 8; ++r) {
            int row = rbase + r;
            if (row < n) Out[(size_t)row * GCN_H + col] = c[t][r];
        }
    }
}

// ---------------- GEMM2: [N,256] x [256,42] -> [N,42] (3 masked 16-col tiles)
__global__ void __launch_bounds__(256) k_gemm2_wmma(const float* __restrict__ A,
                                                    const float* __restrict__ B,
                                                    float* __restrict__ Out,
                                                    int n) {
    const int K = GCN_H, CN = GCN_C;
    int wave = (blockIdx.x * blockDim.x + threadIdx.x) >> 5;
    int lane = threadIdx.x & 31;
    int row0 = wave * 16;
    if (row0 >= n) return;
    int kgrp = lane >> 4;
    int l16  = lane & 15;
    int m = row0 + l16;
    if (m >= n) m = n - 1;
    const float* arow = A + (size_t)m * K;

    v8f z = {0.f,0.f,0.f,0.f,0.f,0.f,0.f,0.f};
    v8f c[3];
    #pragma unroll
    for (int t = 0; t < 3; ++t) c[t] = z;

    for (int k = 0; k < K; k += 4) {        // 64 steps, no remainder
        int kb = k + kgrp * 2;
        v2f a;
        a.x = arow[kb];
        a.y = arow[kb + 1];
        #pragma unroll
        for (int t = 0; t < 3; ++t) {
            int col = t * 16 + l16;
            bool okc = (col < CN);
            v2f b;
            b.x = okc ? B[(size_t)kb * CN + col]       : 0.f;
            b.y = okc ? B[(size_t)(kb + 1) * CN + col] : 0.f;
            c[t] = __builtin_amdgcn_wmma_f32_16x16x4_f32(
                false, a, false, b, (short)0, c[t], false, false);
        }
    }
    int rbase = row0 + kgrp * 8;
    #pragma unroll
    for (int t = 0; t < 3; ++t) {
        int col = t * 16 + l16;
        if (col < CN) {
            #pragma unroll
            for (int r = 0; r < 8; ++r) {
                int row = rbase + r;
                if (row < n) Out[(size_t)row * CN + col] = c[t][r];
            }
        }
    }
}

// ---------------- self-loop init: out[i,f] = h[i,f] * dinv[i]^2 ----------------
__global__ void k_self_init(const float* __restrict__ h, const float* __restrict__ dinv,
                            float* __restrict__ out, int n, int F) {
    long total = (long)n * F;
    long stride = (long)gridDim.x * blockDim.x;
    for (long i = (long)blockIdx.x * blockDim.x + threadIdx.x; i < total; i += stride) {
        int node = (int)(i / F);
        float di = dinv[node];
        out[i] = h[i] * di * di;
    }
}

// ---------------- edge scatter: one wave32 per edge ----------------
__global__ void k_scatter(const int* __restrict__ src, const int* __restrict__ dst,
                          const float* __restrict__ dinv, const float* __restrict__ h,
                          float* __restrict__ out, int e, int F) {
    int lane = threadIdx.x & 31;
    long wave = ((long)blockIdx.x * blockDim.x + threadIdx.x) >> 5;
    long nw = ((long)gridDim.x * blockDim.x) >> 5;
    for (long w = wave; w < e; w += nw) {
        int s = src[w], d = dst[w];
        float nrm = dinv[s] * dinv[d];
        const float* hs = h + (size_t)s * F;
        float* od = out + (size_t)d * F;
        for (int f = lane; f < F; f += 32)
            atomicAdd(&od[f], hs[f] * nrm);
    }
}

// ---------------- bias + relu ----------------
__global__ void k_bias_relu(float* __restrict__ out, const float* __restrict__ bias,
                            int n, int F) {
    long total = (long)n * F;
    long stride = (long)gridDim.x * blockDim.x;
    for (long i = (long)blockIdx.x * blockDim.x + threadIdx.x; i < total; i += stride) {
        int f = (int)(i % F);
        float v = out[i] + bias[f];
        out[i] = v > 0.f ? v : 0.f;
    }
}

// ---------------- bias + log_softmax over C=42, one wave per row ----------------
__global__ void k_logsoftmax(const float* __restrict__ in, const float* __restrict__ bias,
                             float* __restrict__ out, int n, int C) {
    int lane = threadIdx.x & 31;
    long wave = ((long)blockIdx.x * blockDim.x + threadIdx.x) >> 5;
    long nw = ((long)gridDim.x * blockDim.x) >> 5;
    for (long row = wave; row < n; row += nw) {
        const float* rp = in + (size_t)row * C;
        float v0 = (lane < C)      ? rp[lane]      + bias[lane]      : -INFINITY;
        float v1 = (lane + 32 < C) ? rp[lane + 32] + bias[lane + 32] : -INFINITY;
        float mx = fmaxf(v0, v1);
        #pragma unroll
        for (int o = 16; o > 0; o >>= 1) mx = fmaxf(mx, __shfl_xor(mx, o, 32));
        float s = 0.f;
        if (lane < C)      s += expf(v0 - mx);
        if (lane + 32 < C) s += expf(v1 - mx);
        #pragma unroll
        for (int o = 16; o > 0; o >>= 1) s += __shfl_xor(s, o, 32);
        float lse = mx + logf(s);
        float* op = out + (size_t)row * C;
        if (lane < C)      op[lane]      = v0 - lse;
        if (lane + 32 < C) op[lane + 32] = v1 - lse;
    }
}

extern "C" void kernel_launch(void* const* d_in, const int* in_sizes, int n_in,
                              void* d_out, int out_size, void* d_ws, size_t ws_size,
                              hipStream_t stream) {
    const float* x  = (const float*)d_in[0];
    const int*   ei = (const int*)d_in[1];
    const float* W1 = (const float*)d_in[2];
    const float* b1 = (const float*)d_in[3];
    const float* W2 = (const float*)d_in[4];
    const float* b2 = (const float*)d_in[5];

    const int n = in_sizes[0] / GCN_FIN;   // 100000
    const int e = in_sizes[1] / 2;         // 1600000
    const int* src = ei;
    const int* dst = ei + e;

    // workspace layout (floats)
    float* ws   = (float*)d_ws;
    size_t nPad = ((size_t)n + 511) & ~(size_t)511;
    float* dinv = ws;                               // n
    float* h1   = ws + nPad;                        // n * 256
    float* out1 = h1 + (size_t)n * GCN_H;           // n * 256
    float* h2   = out1 + (size_t)n * GCN_H;         // n * 42
    float* out2 = h2 + (size_t)n * GCN_C;           // n * 42
    float* dout = (float*)d_out;

    // symmetric normalization: dinv = (1 + indeg)^-1/2
    k_deg_init<<<(n + 255) / 256, 256, 0, stream>>>(dinv, n);
    k_deg_accum<<<(e + 255) / 256, 256, 0, stream>>>(dst, dinv, e);
    k_deg_rsqrt<<<(n + 255) / 256, 256, 0, stream>>>(dinv, n);

    int waves = (n + 15) / 16;
    int gemmBlocks = (waves * 32 + 255) / 256;

    // layer 1
    k_gemm1_wmma<<<gemmBlocks, 256, 0, stream>>>(x, W1, h1, n);
    k_self_init<<<4096, 256, 0, stream>>>(h1, dinv, out1, n, GCN_H);
    k_scatter<<<8192, 256, 0, stream>>>(src, dst, dinv, h1, out1, e, GCN_H);
    k_bias_relu<<<4096, 256, 0, stream>>>(out1, b1, n, GCN_H);

    // layer 2
    k_gemm2_wmma<<<gemmBlocks, 256, 0, stream>>>(out1, W2, h2, n);
    k_self_init<<<2048, 256, 0, stream>>>(h2, dinv, out2, n, GCN_C);
    k_scatter<<<8192, 256, 0, stream>>>(src, dst, dinv, h2, out2, e, GCN_C);
    k_logsoftmax<<<2048, 256, 0, stream>>>(out2, b2, dout, n, GCN_C);
}